// GCNConv_34626026340408
// MI455X (gfx1250) — compile-verified
//
#include <hip/hip_runtime.h>

#define N_NODES 100000
#define N_EDGES 1600000
#define IN_DIM 128
#define OUT_DIM 64

typedef __attribute__((ext_vector_type(2))) float v2f;
typedef __attribute__((ext_vector_type(8))) float v8f;

// ---------------------------------------------------------------------------
// Kernel 1: h = x @ W using V_WMMA_F32_16X16X4_F32 (fp32 A/B, fp32 acc).
// One wave32 computes a 16-row x 64-col tile of h (4 WMMA accumulators).
// N_NODES (100000) is a multiple of 16, so every tile is full and EXEC is
// all-ones through the WMMA ops (hardware requirement).
//
// VGPR layouts (CDNA5 ISA 7.12.2):
//   A 16x4 f32 : lanes 0-15 -> M=lane, K={0,1}; lanes 16-31 -> M=lane-16, K={2,3}
//   B 4x16 f32 : lanes 0-15 -> N=lane, K={0,1}; lanes 16-31 -> N=lane-16, K={2,3}
//   C/D 16x16  : VGPR v: lanes 0-15 -> (M=v, N=lane); lanes 16-31 -> (M=v+8, N=lane-16)
// ---------------------------------------------------------------------------
__global__ __launch_bounds__(256) void gemm_xw(const float* __restrict__ x,
                                               const float* __restrict__ w,
                                               float* __restrict__ h,
                                               int nRowTiles) {
  const int lane = threadIdx.x & 31;
  const int wave = threadIdx.x >> 5;
  const int rowTile = blockIdx.x * 8 + wave;
  if (rowTile >= nRowTiles) return;        // wave-uniform exit

  const int row0 = rowTile * 16;
  const int m    = lane & 15;              // row (A) / col (B,D) within tile
  const int koff = (lane >> 4) * 2;        // K sub-offset for this half-wave

  const float* xrow = x + (size_t)(row0 + m) * IN_DIM;

  v8f c0 = {}, c1 = {}, c2 = {}, c3 = {};
  for (int k = 0; k < IN_DIM; k += 4) {
    v2f a = *(const v2f*)(xrow + k + koff);          // K=koff, koff+1

    const float* wk0 = w + (size_t)(k + koff) * OUT_DIM + m; // row K=koff
    const float* wk1 = wk0 + OUT_DIM;                        // row K=koff+1
    v2f b0 = { wk0[0],  wk1[0]  };
    v2f b1 = { wk0[16], wk1[16] };
    v2f b2 = { wk0[32], wk1[32] };
    v2f b3 = { wk0[48], wk1[48] };

    c0 = __builtin_amdgcn_wmma_f32_16x16x4_f32(false, a, false, b0, (short)0, c0, false, false);
    c1 = __builtin_amdgcn_wmma_f32_16x16x4_f32(false, a, false, b1, (short)0, c1, false, false);
    c2 = __builtin_amdgcn_wmma_f32_16x16x4_f32(false, a, false, b2, (short)0, c2, false, false);
    c3 = __builtin_amdgcn_wmma_f32_16x16x4_f32(false, a, false, b3, (short)0, c3, false, false);
  }

  const int mOut = (lane < 16) ? 0 : 8;
#pragma unroll
  for (int v = 0; v < 8; ++v) {
    float* hrow = h + (size_t)(row0 + v + mOut) * OUT_DIM + m;
    hrow[0]  = c0[v];
    hrow[16] = c1[v];
    hrow[32] = c2[v];
    hrow[48] = c3[v];
  }
}

// ---------------------------------------------------------------------------
// Kernel 2: out[n][c] = bias[c]  (d_out is poisoned; must initialize)
// ---------------------------------------------------------------------------
__global__ __launch_bounds__(256) void init_out(const float* __restrict__ bias,
                                                float* __restrict__ out,
                                                int total) {
  int i = blockIdx.x * 256 + threadIdx.x;
  if (i < total) out[i] = bias[i & (OUT_DIM - 1)];
}

// ---------------------------------------------------------------------------
// Kernel 3: out[dst] += w_e * h[src].  One wave32 per edge, 2 cols per lane.
// h and out (25.6 MB each) are L2-resident (192 MB L2) -> random gathers and
// fp32 atomics are served by L2, not HBM.
// ---------------------------------------------------------------------------
__global__ __launch_bounds__(256) void scatter_edges(const int* __restrict__ edge_index,
                                                     const float* __restrict__ edge_weight,
                                                     const float* __restrict__ h,
                                                     float* __restrict__ out) {
  const int lane = threadIdx.x & 31;
  const int e = blockIdx.x * 8 + (threadIdx.x >> 5);
  if (e >= N_EDGES) return;

  const int   dst = edge_index[e];            // edge_index[0][e]
  const int   src = edge_index[N_EDGES + e];  // edge_index[1][e]
  const float wgt = edge_weight[e];

  const int c = lane * 2;
  v2f hv = *(const v2f*)(h + (size_t)src * OUT_DIM + c);
  float* op = out + (size_t)dst * OUT_DIM + c;
  atomicAdd(op,     wgt * hv.x);
  atomicAdd(op + 1, wgt * hv.y);
}

extern "C" void kernel_launch(void* const* d_in, const int* in_sizes, int n_in,
                              void* d_out, int out_size, void* d_ws, size_t ws_size,
                              hipStream_t stream) {
  const float* x           = (const float*)d_in[0];
  const int*   edge_index  = (const int*)  d_in[1];
  const float* edge_weight = (const float*)d_in[2];
  const float* weight      = (const float*)d_in[3];
  const float* bias        = (const float*)d_in[4];
  float* out = (float*)d_out;
  float* h   = (float*)d_ws;   // 100000*64*4 = 25.6 MB scratch

  const int nRowTiles = (N_NODES + 15) / 16;                 // 6250 (exact)
  gemm_xw<<<(nRowTiles + 7) / 8, 256, 0, stream>>>(x, weight, h, nRowTiles);

  const int total = N_NODES * OUT_DIM;
  init_out<<<(total + 255) / 256, 256, 0, stream>>>(bias, out, total);

  scatter_edges<<<N_EDGES / 8, 256, 0, stream>>>(edge_index, edge_weight, h, out);
}